// kNNLayer_82454782148637
// MI455X (gfx1250) — compile-verified
//
#include <hip/hip_runtime.h>
#include <hip/hip_bf16.h>

typedef __attribute__((ext_vector_type(2))) float v2f;
typedef __attribute__((ext_vector_type(8))) float v8f;

#define NG    16          // graphs
#define NPER  4096        // points per graph
#define KNN   16          // neighbors
#define TILES (NPER / 16) // 256 column tiles per graph

// Keyed compare-exchange (ascending): branchless, dual-issue friendly.
#define CSWAP(K, I, i, j)                                        \
  {                                                              \
    const bool  sw_ = K[i] > K[j];                               \
    const float ka_ = sw_ ? K[j] : K[i];                         \
    const float kb_ = sw_ ? K[i] : K[j];                         \
    const int   ia_ = sw_ ? I[j] : I[i];                         \
    const int   ib_ = sw_ ? I[i] : I[j];                         \
    K[i] = ka_; K[j] = kb_; I[i] = ia_; I[j] = ib_;              \
  }

// One wave (32 threads) per 16-row tile of one graph.
// V_WMMA_F32_16X16X4_F32 computes each 16x16 gram block (K=4, z-padded D=3);
// LDS transpose hands per-row candidates to all 32 lanes; each lane keeps a
// SORTED top-16 updated per tile with a partial bitonic network
// (sort8 + half-cleaner + bitonic merge16), fully branchless.
__global__ __launch_bounds__(32) void knn_graph_kernel(
    const float* __restrict__ pos, float* __restrict__ out) {
  __shared__ float tileT[16 * 16];  // tileT[col*16 + row] = ||xc||^2 - 2*G
  __shared__ float mD[32][KNN];
  __shared__ int   mI[32][KNN];

  const int lane = threadIdx.x;   // wave32
  const int ln   = lane & 15;
  const int half = lane >> 4;
  const int b    = blockIdx.x / TILES;   // graph id
  const int m    = blockIdx.x % TILES;   // row tile id
  const int graphBase = b * NPER;
  const int rowLocal  = m * 16 + ln;     // this lane's A row (both halves)
  const int rowG      = graphBase + rowLocal;

  // A fragment: 16x4 f32; lanes 0-15 hold K=0,1 in v0,v1; lanes 16-31 K=2,3.
  const float rx = pos[rowG * 3 + 0];
  const float ry = pos[rowG * 3 + 1];
  const float rz = pos[rowG * 3 + 2];
  v2f amat;
  amat.x = half ? rz : rx;
  amat.y = half ? 0.0f : ry;
  const float mySq = rx * rx + ry * ry + rz * rz;

  // Sorted-ascending per-lane top-16 (invariant maintained by the network).
  float key[KNN];
  int   idx[KNN];
#pragma unroll
  for (int t = 0; t < KNN; ++t) { key[t] = INFINITY; idx[t] = 0; }

  // Software-pipelined B-tile position load (hides L2 latency behind merge).
  {
    const int colG0 = graphBase + ln;
    float cx = pos[colG0 * 3 + 0];
    float cy = pos[colG0 * 3 + 1];
    float cz = pos[colG0 * 3 + 2];

    for (int c = 0; c < TILES; ++c) {
      const float bx = cx, by = cy, bz = cz;   // tile c's column position
      // Issue tile c+1's load now; consumed only after this tile's update.
      const int cn    = (c + 1 < TILES) ? (c + 1) : (TILES - 1);
      const int colGn = graphBase + cn * 16 + ln;
      cx = pos[colGn * 3 + 0];
      cy = pos[colGn * 3 + 1];
      cz = pos[colGn * 3 + 2];

      v2f bmat;  // B fragment: 4x16 f32, same per-lane packing as A
      bmat.x = half ? bz : bx;
      bmat.y = half ? 0.0f : by;
      const float sqc = bx * bx + by * by + bz * bz;

      v8f acc = {};
      // D = A(16x4) * B(4x16) + 0 : one 16x16 gram tile per instruction.
      acc = __builtin_amdgcn_wmma_f32_16x16x4_f32(
          false, amat, false, bmat, (short)0, acc, false, false);

      // C layout: VGPR j, lane -> (row = 8*half + j, col = ln).
      // Store transposed with the candidate's ||x||^2 folded in (2x b128).
#pragma unroll
      for (int j = 0; j < 8; ++j) {
        tileT[ln * 16 + half * 8 + j] = sqc - 2.0f * acc[j];
      }
      __syncthreads();

      // Gather this lane's 8 candidates: row ln, cols 8*half .. 8*half+7.
      const int colLocalBase = c * 16;
      float dK[8];
      int   dI[8];
#pragma unroll
      for (int u = 0; u < 8; ++u) {
        const int cc = half * 8 + u;
        dK[u] = mySq + tileT[cc * 16 + ln];
        dI[u] = colLocalBase + cc;
      }

      if (c == m) {  // wave-uniform scalar branch: the only tile with self
#pragma unroll
        for (int u = 0; u < 8; ++u) {
          if (dI[u] == rowLocal) dK[u] = INFINITY;
        }
      }

      float bmin = dK[0];
#pragma unroll
      for (int u = 1; u < 8; ++u) bmin = fminf(bmin, dK[u]);

      if (bmin < key[KNN - 1]) {  // any chance to enter this lane's top-16?
        // (1) Batcher sort8 ascending — 19 comparators.
        CSWAP(dK, dI, 0, 1) CSWAP(dK, dI, 2, 3) CSWAP(dK, dI, 4, 5) CSWAP(dK, dI, 6, 7)
        CSWAP(dK, dI, 0, 2) CSWAP(dK, dI, 1, 3) CSWAP(dK, dI, 4, 6) CSWAP(dK, dI, 5, 7)
        CSWAP(dK, dI, 1, 2) CSWAP(dK, dI, 5, 6)
        CSWAP(dK, dI, 0, 4) CSWAP(dK, dI, 1, 5) CSWAP(dK, dI, 2, 6) CSWAP(dK, dI, 3, 7)
        CSWAP(dK, dI, 2, 4) CSWAP(dK, dI, 3, 5)
        CSWAP(dK, dI, 1, 2) CSWAP(dK, dI, 3, 4) CSWAP(dK, dI, 5, 6)

        // (2) Half-cleaner: [L asc ; B desc-padded] is bitonic(32); low half
        // after pairwise min = the 16 smallest of L∪B, itself bitonic.
#pragma unroll
        for (int i = 8; i < KNN; ++i) {
          const int  j  = 15 - i;               // j = 0..7
          const bool tk = dK[j] < key[i];
          key[i] = tk ? dK[j] : key[i];
          idx[i] = tk ? dI[j] : idx[i];
        }

        // (3) Bitonic merge16 (bitonic -> ascending) — 32 comparators.
#pragma unroll
        for (int i = 0; i < 8; ++i) CSWAP(key, idx, i, i + 8)
#pragma unroll
        for (int h = 0; h < 2; ++h)
#pragma unroll
          for (int i = 0; i < 4; ++i) CSWAP(key, idx, h * 8 + i, h * 8 + i + 4)
#pragma unroll
        for (int q = 0; q < 4; ++q)
#pragma unroll
          for (int i = 0; i < 2; ++i) CSWAP(key, idx, q * 4 + i, q * 4 + i + 2)
#pragma unroll
        for (int p = 0; p < 8; ++p) CSWAP(key, idx, 2 * p, 2 * p + 1)
      }
      __syncthreads();
    }
  }

  // Dump per-lane sorted lists; each row has two half-lists (lanes ln, ln+16).
#pragma unroll
  for (int t = 0; t < KNN; ++t) { mD[lane][t] = key[t]; mI[lane][t] = idx[t]; }
  __syncthreads();

  if (half == 0) {
    // Two-pointer merge of the two sorted 16-lists -> final 16 smallest.
    const int NK = NG * NPER * KNN;    // 1,048,576 edges
    const int e0 = rowG * KNN;
    int pa = 0, pb = 0;
#pragma unroll
    for (int t = 0; t < KNN; ++t) {
      const float da = mD[ln][pa];
      const float db = mD[ln + 16][pb];
      float dsel; int isel;
      if (da <= db) { dsel = da; isel = mI[ln][pa];      ++pa; }
      else          { dsel = db; isel = mI[ln + 16][pb]; ++pb; }
      const int e = e0 + t;
      out[e]          = (float)(graphBase + isel);  // edge_index[0]: neighbor
      out[NK + e]     = (float)rowG;                // edge_index[1]: center
      out[2 * NK + e] = dsel;                       // topk squared distance
    }
  }
}

extern "C" void kernel_launch(void* const* d_in, const int* in_sizes, int n_in,
                              void* d_out, int out_size, void* d_ws, size_t ws_size,
                              hipStream_t stream) {
  (void)in_sizes; (void)n_in; (void)out_size; (void)d_ws; (void)ws_size;
  const float* pos = (const float*)d_in[0];
  // d_in[1] (batch_idx) and d_in[2] (k) encode the fixed 16x4096 / k=16
  // structure, which is baked in (k must be compile-time for the
  // register-resident selection state).
  float* out = (float*)d_out;
  dim3 grid(NG * TILES);  // 4096 single-wave workgroups
  knn_graph_kernel<<<grid, 32, 0, stream>>>(pos, out);
}